// GraphConv_84954453115298
// MI455X (gfx1250) — compile-verified
//
#include <hip/hip_runtime.h>
#include <cstdint>

// Problem constants (match reference)
#define N_USERS    50000
#define N_ENTITIES 100000
#define N_NODES    (N_USERS + N_ENTITIES)
#define D          32
#define N_EDGES    2400000
#define N_HOPS     3
#define KEEP       0.9f
#define INV_KEEP   (1.0f / KEEP)

#define TILE       256   // edges per LDS tile
#define SPMM_BLOCK 256   // 8 wave32 per block

// ---------------------------------------------------------------------------
// gfx1250 async global->LDS copy (ASYNCcnt-tracked DMA path).
// lds_addr: LDS byte offset (low 32 bits of a generic pointer to __shared__).
// voff: per-lane byte offset VGPR. sbase: uniform 64-bit base in SGPR pair.
// Edge stream is single-use per hop -> non-temporal hint keeps the hot node
// working set (x/y/agg, ~58 MB) resident in the 192 MB L2.
// ---------------------------------------------------------------------------
__device__ __forceinline__ void async_b32_to_lds(unsigned lds_addr, unsigned voff,
                                                 uint64_t sbase) {
  asm volatile("global_load_async_to_lds_b32 %0, %1, %2 th:TH_LOAD_NT"
               :: "v"(lds_addr), "v"(voff), "s"(sbase) : "memory");
}

__device__ __forceinline__ void wait_async_le3() {
  asm volatile("s_wait_asynccnt 0x3" ::: "memory");
}
__device__ __forceinline__ void wait_async_le0() {
  asm volatile("s_wait_asynccnt 0x0" ::: "memory");
}

// ---------------------------------------------------------------------------
// Kernel 1: agg = cur = concat(user_emb, entity_emb)
// ---------------------------------------------------------------------------
__global__ void init_kernel(const float4* __restrict__ u,
                            const float4* __restrict__ e,
                            float4* __restrict__ agg,
                            float4* __restrict__ cur) {
  const int NU4 = N_USERS * D / 4;
  const int NT4 = N_NODES * D / 4;
  int i = blockIdx.x * blockDim.x + threadIdx.x;
  if (i >= NT4) return;
  float4 v = (i < NU4) ? u[i] : e[i - NU4];
  agg[i] = v;
  cur[i] = v;
}

// ---------------------------------------------------------------------------
// Kernel 2: zero the scatter target (only once, before hop 0; afterwards
// drop_acc re-zeroes nxt in the same pass that consumes it)
// ---------------------------------------------------------------------------
__global__ void zero_kernel(float4* __restrict__ p, int n4) {
  int i = blockIdx.x * blockDim.x + threadIdx.x;
  if (i < n4) p[i] = make_float4(0.f, 0.f, 0.f, 0.f);
}

// ---------------------------------------------------------------------------
// Kernel 3: SpMM  y[head] += val * x[tail], unsorted edges, f32 atomics.
// Edge indices/vals double-buffered through LDS via the async DMA path.
// ---------------------------------------------------------------------------
__global__ __launch_bounds__(SPMM_BLOCK)
void spmm_kernel(const long long* __restrict__ head,
                 const long long* __restrict__ tail,
                 const float* __restrict__ vals,
                 const float* __restrict__ x,
                 float* __restrict__ y) {
  __shared__ int   sh_head[2][TILE];
  __shared__ int   sh_tail[2][TILE];
  __shared__ float sh_val [2][TILE];

  const int tid    = threadIdx.x;
  const int nTiles = (N_EDGES + TILE - 1) / TILE;
  const uint64_t hb = (uint64_t)head;  // int64 in memory; DMA the low DWORD
  const uint64_t tb = (uint64_t)tail;
  const uint64_t vb = (uint64_t)vals;

  int t = (int)blockIdx.x;
  if (t >= nTiles) return;

  auto issue = [&](int tile, int buf) {
    long long e = (long long)tile * TILE + tid;
    if (e >= N_EDGES) e = N_EDGES - 1;           // clamp: redundant, guarded later
    unsigned o8 = (unsigned)(e * 8);             // int64 stride
    unsigned o4 = (unsigned)(e * 4);
    async_b32_to_lds((unsigned)(uintptr_t)&sh_head[buf][tid], o8, hb);
    async_b32_to_lds((unsigned)(uintptr_t)&sh_tail[buf][tid], o8, tb);
    async_b32_to_lds((unsigned)(uintptr_t)&sh_val [buf][tid], o4, vb);
  };

  int buf = 0;
  issue(t, 0);

  for (; t < nTiles; t += (int)gridDim.x) {
    int nt = t + (int)gridDim.x;
    if (nt < nTiles) {
      issue(nt, buf ^ 1);      // prefetch next tile into the other buffer
      wait_async_le3();        // retire only this tile's 3 async ops per wave
    } else {
      wait_async_le0();
    }
    __syncthreads();           // LDS tile visible to all 8 waves

    const int sub   = tid & 7;       // float4 slice: 8 lanes cover D=32
    const int slot0 = tid >> 3;      // 32 edges per pass, 8 passes per tile
    const long long ebase = (long long)t * TILE;
#pragma unroll
    for (int k = 0; k < TILE / 32; ++k) {
      int slot = slot0 + k * 32;
      if (ebase + slot < N_EDGES) {
        int   h  = sh_head[buf][slot];
        int   tl = sh_tail[buf][slot];
        float v  = sh_val [buf][slot];
        float4 g = ((const float4*)(x + (size_t)tl * D))[sub];   // global_load_b128
        float* yr = y + (size_t)h * D + sub * 4;
        atomicAdd(yr + 0, v * g.x);  // global_atomic_add_f32, non-returning
        atomicAdd(yr + 1, v * g.y);
        atomicAdd(yr + 2, v * g.z);
        atomicAdd(yr + 3, v * g.w);
      }
    }
    __syncthreads();           // all waves done reading buf before it is reused
    buf ^= 1;
  }
}

// ---------------------------------------------------------------------------
// Kernel 4: dropout + layer accumulate + re-zero nxt for the next hop.
// Deterministic counter-based PCG hash keyed on (hop, element); same
// keep-prob / 1/keep scaling semantics as nn.Dropout (bitwise JAX threefry
// reproduction is not attempted in this compile-only environment).
// ---------------------------------------------------------------------------
__device__ __forceinline__ unsigned pcg_hash(unsigned v) {
  unsigned s = v * 747796405u + 2891336453u;
  unsigned w = ((s >> ((s >> 28u) + 4u)) ^ s) * 277803737u;
  return (w >> 22u) ^ w;
}

__global__ void drop_acc_kernel(float* __restrict__ nxt,
                                float* __restrict__ cur,
                                float* __restrict__ agg,
                                unsigned hop_seed) {
  int i = blockIdx.x * blockDim.x + threadIdx.x;
  if (i >= N_NODES * D) return;
  unsigned r = pcg_hash((unsigned)i * 0x9E3779B9u + hop_seed * 0x85EBCA6Bu);
  float u = (float)(r >> 8) * (1.0f / 16777216.0f);     // [0,1)
  float s = __builtin_nontemporal_load(nxt + i);        // last-use read
  float v = (u < KEEP) ? s * INV_KEEP : 0.0f;
  nxt[i] = 0.0f;                                        // ready for next scatter
  cur[i] = v;
  agg[i] += v;
}

// ---------------------------------------------------------------------------
// Host-side launch
// ---------------------------------------------------------------------------
extern "C" void kernel_launch(void* const* d_in, const int* in_sizes, int n_in,
                              void* d_out, int out_size, void* d_ws, size_t ws_size,
                              hipStream_t stream) {
  (void)in_sizes; (void)n_in; (void)out_size; (void)ws_size;

  const float*     user_emb = (const float*)d_in[0];
  const float*     ent_emb  = (const float*)d_in[1];
  const long long* eh       = (const long long*)d_in[2];   // int64
  const long long* et       = (const long long*)d_in[3];   // int64
  const float*     ev       = (const float*)d_in[4];

  float* agg = (float*)d_out;                    // [N_NODES, D] layer sum
  float* cur = (float*)d_ws;                     // [N_NODES, D]
  float* nxt = cur + (size_t)N_NODES * D;        // [N_NODES, D]

  const int n4 = N_NODES * D / 4;
  const int nd = N_NODES * D;
  const int nTiles = (N_EDGES + TILE - 1) / TILE;
  const int spmmBlocks = nTiles < 2048 ? nTiles : 2048;

  init_kernel<<<(n4 + 255) / 256, 256, 0, stream>>>(
      (const float4*)user_emb, (const float4*)ent_emb,
      (float4*)agg, (float4*)cur);
  zero_kernel<<<(n4 + 255) / 256, 256, 0, stream>>>((float4*)nxt, n4);

  for (int hop = 0; hop < N_HOPS; ++hop) {
    spmm_kernel<<<spmmBlocks, SPMM_BLOCK, 0, stream>>>(eh, et, ev, cur, nxt);
    drop_acc_kernel<<<(nd + 255) / 256, 256, 0, stream>>>(
        nxt, cur, agg, (unsigned)(hop + 1));
  }
}